// InstanceHead_69166153335044
// MI455X (gfx1250) — compile-verified
//
#include <hip/hip_runtime.h>
#include <math.h>

// Problem constants (from reference): N=100000, M=256, B=8, L=128, D=64
#define N_PTS   100000
#define M_CEN   256
#define L_IN    128
#define D_OUT   64
#define TILE    16
#define NTILES  (N_PTS / TILE)   // 6250, exact
#define THREADS 128              // 4 waves (wave32)

typedef float v8f   __attribute__((ext_vector_type(8)));
typedef float v2f   __attribute__((ext_vector_type(2)));
typedef float f32x4 __attribute__((ext_vector_type(4)));
typedef int   i32x4 __attribute__((ext_vector_type(4)));

// ---- CDNA5 async global->LDS path (ASYNCcnt-tracked DMA), with fallback ----
#if defined(__AMDGCN__) && __has_builtin(__builtin_amdgcn_global_load_async_to_lds_b128)
#define ASYNC_LDS_COPY 1
typedef int v4i_vs __attribute__((vector_size(16)));              // matches builtin param
typedef __attribute__((address_space(1))) v4i_vs* gptr_b128;      // global v4i*
typedef __attribute__((address_space(3))) v4i_vs* lptr_b128;      // LDS v4i*
#define ASYNC_CP_B128(gsrc, ldst) \
  __builtin_amdgcn_global_load_async_to_lds_b128((gptr_b128)(gsrc), (lptr_b128)(ldst), 0, 0)
#if __has_builtin(__builtin_amdgcn_s_wait_asynccnt)
#define WAIT_ASYNC(n) __builtin_amdgcn_s_wait_asynccnt(n)
#else
#define WAIT_ASYNC(n) asm volatile("s_wait_asynccnt %0" :: "i"(n) : "memory")
#endif
#else
#define ASYNC_LDS_COPY 0
#define ASYNC_CP_B128(gsrc, ldst) (*(f32x4*)(ldst) = *(const f32x4*)(gsrc))
#define WAIT_ASYNC(n) ((void)0)
#endif

// ---------------------------------------------------------------------------
// Kernel 1: cen[m][d] = conf[m] * (centroid_feats[m,:] . W[:,d] + b[d])
// Stored K-pair interleaved: cenP[(d/2)][m][d&1]  (i.e. [32][256][2] floats)
// so a WMMA lane's two K values are one aligned b64 in LDS.
// ---------------------------------------------------------------------------
__global__ void cen_prep_kernel(const float* __restrict__ cf,
                                const float* __restrict__ conf,
                                const float* __restrict__ W,
                                const float* __restrict__ b,
                                float* __restrict__ cenP) {
  const int d = blockIdx.x;    // 0..63
  const int m = threadIdx.x;   // 0..255
  float acc = b[d];
  for (int k = 0; k < L_IN; ++k)
    acc = fmaf(cf[m * L_IN + k], W[k * D_OUT + d], acc);
  cenP[(d >> 1) * (2 * M_CEN) + 2 * m + (d & 1)] = conf[m] * acc;
}

// ---------------------------------------------------------------------------
// Main fused kernel: per 16-row tile
//   GEMM1 (feats@W+b) -> row-normalize -> masked-softmax attn -> GEMM2
//   (clu@cenT) -> *attn -> store.   All GEMMs via V_WMMA_F32_16X16X4_F32.
//   Feature tiles double-buffered via GLOBAL_LOAD_ASYNC_TO_LDS (ASYNCcnt).
// ---------------------------------------------------------------------------
__global__ __launch_bounds__(THREADS)
void instance_head_kernel(const int*   __restrict__ clu_coords,   // [N,4]
                          const int*   __restrict__ cen_coords,   // [M,4]
                          const float* __restrict__ feats,        // [N,128]
                          const float* __restrict__ W,            // [128,64]
                          const float* __restrict__ b,            // [64]
                          const float* __restrict__ cenP,         // [32][256][2]
                          float*       __restrict__ out) {        // [N,256]
  __shared__ __align__(16) float sCen[D_OUT * M_CEN];    // 64 KB : paired B-matrix
  __shared__ __align__(16) float sFeat[2][TILE * L_IN];  // 16 KB : dbl-buffered A
  __shared__ __align__(16) float sClu[TILE * D_OUT];     //  4 KB : GEMM1 out
  __shared__ __align__(16) float sAttn[TILE * M_CEN];    // 16 KB : softmax wgts
  __shared__ __align__(16) int   sCenC[M_CEN * 4];       //  4 KB : centroid coords
  __shared__ float sInv[TILE];                           // 1/row-norm

  const int tid  = threadIdx.x;
  const int wave = tid >> 5;
  const int lane = tid & 31;
  const int half = lane >> 4;   // 0: lanes 0-15, 1: lanes 16-31
  const int l16  = lane & 15;

  // ---- block init: async-stage cenP (64KB, 32 ops/wave) + coords (2 ops) --
  for (int i = tid; i < (D_OUT * M_CEN) / 4; i += THREADS)
    ASYNC_CP_B128((const f32x4*)cenP + i, (f32x4*)sCen + i);
  for (int i = tid; i < M_CEN; i += THREADS)
    ASYNC_CP_B128((const i32x4*)cen_coords + i, (i32x4*)sCenC + i);

  // ---- per-wave: preload W B-fragments for GEMM1 (16 output cols) --------
  // B (4x16 f32) layout: vgpr0 = K {0|2} (low|high lane half), vgpr1 = K {1|3}
  const int dBase = wave * 16;
  v2f wf[32];
#pragma unroll
  for (int kt = 0; kt < 32; ++kt) {
    const int k = 4 * kt + 2 * half;
    wf[kt].x = W[k * D_OUT + dBase + l16];
    wf[kt].y = W[(k + 1) * D_OUT + dBase + l16];
  }
  const float bias = b[dBase + l16];

  // ---- prologue: async-prefetch first feature tile into buffer 0 ---------
  int tile = blockIdx.x;
  if (tile < NTILES) {
#pragma unroll
    for (int s = 0; s < 4; ++s) {
      const int idx = tid + s * THREADS;  // float4 index within 16x128 tile
      ASYNC_CP_B128((const f32x4*)(feats + (size_t)tile * TILE * L_IN) + idx,
                    (f32x4*)sFeat[0] + idx);
    }
  }

  int buf = 0;
  for (; tile < NTILES; tile += gridDim.x, buf ^= 1) {
    const int n0 = tile * TILE;

    // ---- async-prefetch next tile into the other buffer -----------------
    const int nextTile = tile + gridDim.x;
    const bool hasNext = (nextTile < NTILES);
    if (hasNext) {
#pragma unroll
      for (int s = 0; s < 4; ++s) {
        const int idx = tid + s * THREADS;
        ASYNC_CP_B128((const f32x4*)(feats + (size_t)nextTile * TILE * L_IN) + idx,
                      (f32x4*)sFeat[buf ^ 1] + idx);
      }
      WAIT_ASYNC(4);   // in-order completion: current buffer (+init) retired
    } else {
      WAIT_ASYNC(0);
    }
    __syncthreads();   // LDS visibility of async DMA across all 4 waves

    const float* __restrict__ ft = sFeat[buf];

    // ---- GEMM1: [16x128] @ [128x16] + b, this wave's 16 columns ---------
    v8f acc;
#pragma unroll
    for (int v = 0; v < 8; ++v) acc[v] = bias;  // C/D: col = l16, row = v+8*half
#pragma unroll
    for (int kt = 0; kt < 32; ++kt) {
      const int k = 4 * kt + 2 * half;          // A: row = l16, K split by half
      const v2f a = *(const v2f*)&ft[l16 * L_IN + k];   // one ds_load_b64
      acc = __builtin_amdgcn_wmma_f32_16x16x4_f32(
          /*neg_a=*/false, a, /*neg_b=*/false, wf[kt],
          /*c_mod=*/(short)0, acc, /*reuse_a=*/false, /*reuse_b=*/false);
    }
#pragma unroll
    for (int v = 0; v < 8; ++v)
      sClu[(v + 8 * half) * D_OUT + dBase + l16] = acc[v];

    // ---- attention weights: this wave owns rows wave*4 .. wave*4+3 -------
    for (int rr = 0; rr < 4; ++rr) {
      const int r  = wave * 4 + rr;
      const int cb = clu_coords[(n0 + r) * 4 + 0];
      const float cx = (float)clu_coords[(n0 + r) * 4 + 1];
      const float cy = (float)clu_coords[(n0 + r) * 4 + 2];
      const float cz = (float)clu_coords[(n0 + r) * 4 + 3];
      float lg[8];
      int   vd[8];
      float mx = -__builtin_inff();
#pragma unroll
      for (int j = 0; j < 8; ++j) {
        const int m = lane + 32 * j;
        const int eb = sCenC[m * 4 + 0];
        const float dx = cx - (float)sCenC[m * 4 + 1];
        const float dy = cy - (float)sCenC[m * 4 + 2];
        const float dz = cz - (float)sCenC[m * 4 + 3];
        const float d2 = fmaf(dx, dx, fmaf(dy, dy, dz * dz));
        lg[j] = -fmaxf(sqrtf(d2), 0.1f);        // logits = -clamped dist
        vd[j] = (eb == cb);
        if (vd[j]) mx = fmaxf(mx, lg[j]);
      }
#pragma unroll
      for (int o = 16; o > 0; o >>= 1) mx = fmaxf(mx, __shfl_xor(mx, o, 32));
      float e[8];
      float s = 0.f;
#pragma unroll
      for (int j = 0; j < 8; ++j) {
        e[j] = vd[j] ? __expf(lg[j] - mx) : 0.f;
        s += e[j];
      }
#pragma unroll
      for (int o = 16; o > 0; o >>= 1) s += __shfl_xor(s, o, 32);
      const float inv = (s > 0.f) ? (1.f / s) : 0.f;  // empty row -> attn 0
#pragma unroll
      for (int j = 0; j < 8; ++j)
        sAttn[r * M_CEN + lane + 32 * j] = e[j] * inv;
    }
    __syncthreads();

    // ---- row norms: nrm = max(||clu||, 1e-12) ---------------------------
    if (tid < TILE) {
      float s = 0.f;
      for (int d = 0; d < D_OUT; ++d) {
        const float v = sClu[tid * D_OUT + d];
        s = fmaf(v, v, s);
      }
      sInv[tid] = 1.0f / fmaxf(sqrtf(s), 1e-12f);
    }
    __syncthreads();

    // ---- GEMM2: [16x64] @ [64x64] (this wave's 64 of 256 columns) -------
    const int mBase = wave * 64;
    const float sc = sInv[l16];                 // A row = l16; normalization
    v8f c2[4];
#pragma unroll
    for (int t = 0; t < 4; ++t)
#pragma unroll
      for (int v = 0; v < 8; ++v) c2[t][v] = 0.f;
#pragma unroll
    for (int ks = 0; ks < 16; ++ks) {
      const int k = 4 * ks + 2 * half;
      v2f a = *(const v2f*)&sClu[l16 * D_OUT + k];     // one ds_load_b64
      a.x *= sc;
      a.y *= sc;
#pragma unroll
      for (int t = 0; t < 4; ++t) {
        const int col = mBase + 16 * t + l16;
        // paired layout: (cen[k][col], cen[k+1][col]) adjacent -> one b64
        const v2f bf =
            *(const v2f*)&sCen[(2 * ks + half) * (2 * M_CEN) + 2 * col];
        c2[t] = __builtin_amdgcn_wmma_f32_16x16x4_f32(
            false, a, false, bf, (short)0, c2[t], false, false);
      }
    }

    // ---- scale by attn and store (two 64B runs per b32 store) -----------
#pragma unroll
    for (int t = 0; t < 4; ++t) {
      const int col = mBase + 16 * t + l16;
#pragma unroll
      for (int v = 0; v < 8; ++v) {
        const int row = v + 8 * half;
        const float wgt = sAttn[row * M_CEN + col];
        out[(size_t)(n0 + row) * M_CEN + col] = c2[t][v] * wgt;
      }
    }
    __syncthreads();   // protect sClu/sAttn (and allow next prefetch reuse)
  }
}

// ---------------------------------------------------------------------------
extern "C" void kernel_launch(void* const* d_in, const int* in_sizes, int n_in,
                              void* d_out, int out_size, void* d_ws, size_t ws_size,
                              hipStream_t stream) {
  (void)in_sizes; (void)n_in; (void)out_size; (void)ws_size;
  const int*   clu_coords = (const int*)  d_in[0];  // [N,4]
  const int*   cen_coords = (const int*)  d_in[1];  // [M,4]
  const float* clu_feats  = (const float*)d_in[2];  // [N,128]
  const float* cen_feats  = (const float*)d_in[3];  // [M,128]
  const float* conf       = (const float*)d_in[4];  // [M,1]
  const float* W          = (const float*)d_in[5];  // [128,64]
  const float* b          = (const float*)d_in[6];  // [64]
  float* out  = (float*)d_out;                      // [N,256]
  float* cenP = (float*)d_ws;                       // 64 KB scratch (paired)

  cen_prep_kernel<<<D_OUT, M_CEN, 0, stream>>>(cen_feats, conf, W, b, cenP);
  instance_head_kernel<<<2048, THREADS, 0, stream>>>(
      clu_coords, cen_coords, clu_feats, W, b, cenP, out);
}